// ContrastEnergy_83133386981512
// MI455X (gfx1250) — compile-verified
//
#include <hip/hip_runtime.h>
#include <math.h>

// CDNA5 / gfx1250: wave32, V_WMMA_F32_16X16X4_F32 for the affine x stencil einsum.
// Stencils staged in LDS; image gathers use 32-bit offsets (SADDR form).

typedef __attribute__((ext_vector_type(2))) float v2f;
typedef __attribute__((ext_vector_type(8))) float v8f;

#define IMG_H 2048
#define IMG_W 2048
#define EPSC  1e-8f
#define MAX_IN  1024   // 32x32 interior stencil
#define MAX_OUT 256    // ring stencil (124) padded to tile multiple

__device__ __forceinline__ float bilin(const float* __restrict__ img, float vi, float vj) {
    // clamped bilinear, mirrors reference exactly
    float fi = floorf(vi);
    fi = fminf(fmaxf(fi, 0.0f), (float)(IMG_H - 2));
    float fj = floorf(vj);
    fj = fminf(fmaxf(fj, 0.0f), (float)(IMG_W - 2));
    const uint32_t i0 = (uint32_t)(int)fi;       // in [0, 2046]
    const uint32_t j0 = (uint32_t)(int)fj;       // in [0, 2046]
    const float di = fminf(fmaxf(vi - fi, 0.0f), 1.0f);
    const float dj = fminf(fmaxf(vj - fj, 0.0f), 1.0f);
    // 32-bit linear index: 2048*2048*4 bytes = 16MB < 2^31 -> SADDR addressing
    const float* p = img + (i0 * (uint32_t)IMG_W + j0);
    const float v00 = p[0];
    const float v01 = p[1];
    const float v10 = p[IMG_W];
    const float v11 = p[IMG_W + 1];
    return v00 * (1.0f - di) * (1.0f - dj) + v01 * (1.0f - di) * dj
         + v10 * di * (1.0f - dj)          + v11 * di * dj;
}

__device__ __forceinline__ v8f coord_wmma(v2f A, v2f B) {
    v8f cz = {};
    return __builtin_amdgcn_wmma_f32_16x16x4_f32(
        false, A, false, B, (short)0, cz, false, false);
}

extern "C" __global__ __launch_bounds__(256)
void contrast_energy_kernel(const float* __restrict__ points,   // (S*P, 5)
                            const float* __restrict__ pmask,    // (S*P)
                            const float* __restrict__ img,      // (H, W)
                            const float* __restrict__ st_in, int n_in,
                            const float* __restrict__ st_out, int n_out,
                            float* __restrict__ out, int total_pts)
{
    __shared__ float2 shin[MAX_IN];
    __shared__ float2 shout[MAX_OUT];

    const int tid  = (int)threadIdx.x;
    const int lane = tid & 31;
    const int wave = tid >> 5;
    const int wavesPerBlock = (int)(blockDim.x >> 5);
    const int tile = (int)blockIdx.x * wavesPerBlock + wave;
    const int p0 = tile * 16;                 // this wave owns points p0..p0+15

    // ---- cooperative stencil staging into LDS (all threads participate) ----
    for (int i = tid; i < n_in; i += (int)blockDim.x) {
        shin[i] = make_float2(st_in[3 * i], st_in[3 * i + 1]);
    }
    const int nout_pad = (n_out + 15) & ~15;
    for (int i = tid; i < nout_pad; i += (int)blockDim.x) {
        const int c = (i < n_out) ? i : (n_out - 1);
        shout[i] = make_float2(st_out[3 * c], st_out[3 * c + 1]);
    }
    __syncthreads();

    if (p0 >= total_pts) return;              // wave-uniform; EXEC stays all-1s

    const bool hi   = lane >= 16;
    const int  msub = lane & 15;

    // A fragments, 16x4 f32 layout: lanes 0-15 carry K=0,1; lanes 16-31 carry K=2,3.
    const float* st = points + (size_t)(p0 + msub) * 5;
    const float px = st[0], py = st[1], pa = st[2], pb = st[3], pt = st[4];
    float sn, cs;
    sincosf(pt, &sn, &cs);

    v2f Ai, Aj;                               // row-i: (a c, -b s, x) ; row-j: (a s, b c, y)
    Ai.x = hi ? px   : pa * cs;
    Ai.y = hi ? 0.0f : -pb * sn;
    Aj.x = hi ? py   : pa * sn;
    Aj.y = hi ? 0.0f : pb * cs;

    float s_in[8]  = {0.f, 0.f, 0.f, 0.f, 0.f, 0.f, 0.f, 0.f};
    float q_in[8]  = {0.f, 0.f, 0.f, 0.f, 0.f, 0.f, 0.f, 0.f};
    float s_out[8] = {0.f, 0.f, 0.f, 0.f, 0.f, 0.f, 0.f, 0.f};
    float q_out[8] = {0.f, 0.f, 0.f, 0.f, 0.f, 0.f, 0.f, 0.f};

    // ---- interior stencil: all tiles dense, branch-free ----
    const int full_in = n_in >> 4;            // 64
#pragma unroll 2
    for (int ti = 0; ti < full_in; ++ti) {
        const float2 g = shin[(ti << 4) + msub];
        v2f B;                                 // 4x16: lanes 0-15 = (gx,gy); 16-31 = (1,0)
        B.x = hi ? 1.0f : g.x;
        B.y = hi ? 0.0f : g.y;
        const v8f Di = coord_wmma(Ai, B);
        const v8f Dj = coord_wmma(Aj, B);
#pragma unroll
        for (int r = 0; r < 8; ++r) {
            const float v = bilin(img, Di[r], Dj[r]);
            s_in[r] += v;
            q_in[r] += v * v;
        }
    }

    // ---- ring stencil: dense tiles, then one ragged tail tile ----
    const int full_out = n_out >> 4;          // 7
    for (int ti = 0; ti < full_out; ++ti) {
        const float2 g = shout[(ti << 4) + msub];
        v2f B;
        B.x = hi ? 1.0f : g.x;
        B.y = hi ? 0.0f : g.y;
        const v8f Di = coord_wmma(Ai, B);
        const v8f Dj = coord_wmma(Aj, B);
#pragma unroll
        for (int r = 0; r < 8; ++r) {
            const float v = bilin(img, Di[r], Dj[r]);
            s_out[r] += v;
            q_out[r] += v * v;
        }
    }
    const int tail = n_out & 15;              // 12
    if (tail) {                               // uniform branch; WMMA sees EXEC all-1s
        const float2 g = shout[(full_out << 4) + msub];
        v2f B;
        B.x = hi ? 1.0f : g.x;
        B.y = hi ? 0.0f : g.y;
        const v8f Di = coord_wmma(Ai, B);
        const v8f Dj = coord_wmma(Aj, B);
        if (msub < tail) {
#pragma unroll
            for (int r = 0; r < 8; ++r) {
                const float v = bilin(img, Di[r], Dj[r]);
                s_out[r] += v;
                q_out[r] += v * v;
            }
        }
    }

    // ---- reduce over stencil (N) dim: butterfly within each 16-lane half ----
#pragma unroll
    for (int r = 0; r < 8; ++r) {
#pragma unroll
        for (int off = 1; off < 16; off <<= 1) {
            s_in[r]  += __shfl_xor(s_in[r],  off, 32);
            q_in[r]  += __shfl_xor(q_in[r],  off, 32);
            s_out[r] += __shfl_xor(s_out[r], off, 32);
            q_out[r] += __shfl_xor(q_out[r], off, 32);
        }
    }

    if (msub == 0) {
        const int   mbase = hi ? 8 : 0;
        const float nin  = (float)n_in;
        const float nout = (float)n_out;
#pragma unroll
        for (int r = 0; r < 8; ++r) {
            const float mi = s_in[r] / nin;
            const float vi = (q_in[r]  - nin  * mi * mi) / (nin  - 1.0f);  // ddof=1
            const float mo = s_out[r] / nout;
            const float vo = (q_out[r] - nout * mo * mo) / (nout - 1.0f);
            const int pidx = p0 + mbase + r;
            out[pidx] = (mi - mo) / sqrtf(vi + vo + EPSC) * pmask[pidx];
        }
    }
}

extern "C" void kernel_launch(void* const* d_in, const int* in_sizes, int n_in_arrs,
                              void* d_out, int out_size, void* d_ws, size_t ws_size,
                              hipStream_t stream) {
    (void)n_in_arrs; (void)out_size; (void)d_ws; (void)ws_size;
    const float* points = (const float*)d_in[0];   // (S, P, 5)   float32
    const float* pmask  = (const float*)d_in[1];   // (S, P)      float32
    const float* img    = (const float*)d_in[2];   // (1, H, W)   float32
    const float* st_in  = (const float*)d_in[3];   // (n_in, 3)   float32
    const float* st_out = (const float*)d_in[4];   // (n_out, 3)  float32
    float* out = (float*)d_out;                    // (S, P)      float32

    const int total  = in_sizes[1];        // S*P = 16384
    const int nin    = in_sizes[3] / 3;    // 1024
    const int nout   = in_sizes[4] / 3;    // 124

    const int waves          = (total + 15) / 16;   // 16 points per wave
    const int threads        = 256;                 // 8 waves / block (wave32)
    const int wavesPerBlock  = threads / 32;
    const int blocks         = (waves + wavesPerBlock - 1) / wavesPerBlock;

    contrast_energy_kernel<<<blocks, threads, 0, stream>>>(
        points, pmask, img, st_in, nin, st_out, nout, out, total);
}